// PointLoss_46016279609389
// MI455X (gfx1250) — compile-verified
//
#include <hip/hip_runtime.h>
#include <hip/hip_bf16.h>
#include <math.h>

typedef float v2f __attribute__((ext_vector_type(2)));
typedef float v8f __attribute__((ext_vector_type(8)));

#define NB 8
#define NPTS 4096
#define ST 64                         // super-tile edge (points)
#define NST (NPTS / ST)               // 64 super-tiles per dimension
#define SUPER_UT ((NST * (NST + 1)) / 2)   // 2080 upper-tri super-tiles
#define H 512
#define W 512
#define KS 11
#define PAD 5
#define MINDIST_C 0.05f
#define MAXVAL_C 255.0f

// gaussian kernel, sigma = 0.3*((11-1)*0.5 - 1) + 0.8 = 2.0, normalized
__constant__ float KW[KS] = {
    0.00881223f, 0.02714357f, 0.06511404f, 0.12164908f, 0.17699836f,
    0.20056542f,
    0.17699836f, 0.12164908f, 0.06511404f, 0.02714357f, 0.00881223f
};

__device__ __forceinline__ int refl511(int k) {
    k = (k < 0) ? -k : k;
    return (k > 511) ? (1022 - k) : k;
}

// ---------------- accumulator init ----------------
__global__ void init_acc_kernel(float* acc) {
    if (threadIdx.x < 2) acc[threadIdx.x] = 0.0f;
}

// ---------------- pairwise hinge via WMMA Gram tiles ----------------
// One 512-thread block (16 waves) = one 64x64 super-tile of the pairwise
// matrix. Points staged in LDS; each wave computes one 16x16 Gram tile
// with v_wmma_f32_16x16x4_f32, then the hinge epilogue.
__global__ void __launch_bounds__(512)
dist_wmma_kernel(const float* __restrict__ trace,
                 float* __restrict__ acc) {
    const int gt = blockIdx.x;                 // 0 .. NB*SUPER_UT-1
    const int batch = gt / SUPER_UT;
    const int t = gt - batch * SUPER_UT;

    // invert linear upper-triangle index (incl. diagonal) -> (Ti, Tj), T=64
    float disc = (float)((2 * NST + 1) * (2 * NST + 1) - 8 * t);
    int Ti = (int)(((float)(2 * NST + 1) - sqrtf(disc)) * 0.5f);
    if (Ti < 0) Ti = 0;
    if (Ti > NST - 1) Ti = NST - 1;
    while (Ti > 0 && (Ti * NST - (Ti * (Ti - 1)) / 2) > t) --Ti;
    while (((Ti + 1) * NST - ((Ti + 1) * Ti) / 2) <= t) ++Ti;
    const int Tj = Ti + (t - (Ti * NST - (Ti * (Ti - 1)) / 2));

    const float* tr = trace + (size_t)batch * NPTS * 2;

    __shared__ float rowx[ST], rowy[ST], rowsq[ST];
    __shared__ float colx[ST], coly[ST], colsq[ST];
    __shared__ float wavesum[16];

    const int tid = threadIdx.x;
    __builtin_prefetch(&tr[(size_t)(Tj * ST) * 2], 0, 0);   // global_prefetch_b8
    if (tid < ST) {
        float x = tr[(size_t)(Ti * ST + tid) * 2 + 0];
        float y = tr[(size_t)(Ti * ST + tid) * 2 + 1];
        rowx[tid] = x; rowy[tid] = y; rowsq[tid] = x * x + y * y;
    } else if (tid < 2 * ST) {
        int k = tid - ST;
        float x = tr[(size_t)(Tj * ST + k) * 2 + 0];
        float y = tr[(size_t)(Tj * ST + k) * 2 + 1];
        colx[k] = x; coly[k] = y; colsq[k] = x * x + y * y;
    }
    __syncthreads();

    const int wave = tid >> 5;
    const int lane = tid & 31;
    const int ri = (wave >> 2) << 4;           // row sub-tile offset (0,16,32,48)
    const int ci = (wave & 3) << 4;            // col sub-tile offset
    const int m = lane & 15;

    // A: 16x4 f32 (rows x {x,y,0,0}); lanes 0-15 = K0,K1; lanes 16-31 = K2,K3 (zero)
    v2f a = {0.0f, 0.0f};
    v2f b = {0.0f, 0.0f};
    if (lane < 16) {
        a[0] = rowx[ri + m]; a[1] = rowy[ri + m];
        b[0] = colx[ci + m]; b[1] = coly[ci + m];
    }
    v8f g = {};
    g = __builtin_amdgcn_wmma_f32_16x16x4_f32(
            /*neg_a=*/false, a, /*neg_b=*/false, b,
            /*c_mod=*/(short)0, g, /*reuse_a=*/false, /*reuse_b=*/false);

    // D layout: VGPR r -> row M=r (lanes 0-15) / M=r+8 (lanes 16-31); col N = lane&15
    const int n = lane & 15;
    const int jg = Tj * ST + ci + n;           // global column point index
    const float sqj = colsq[ci + n];
    const int rowBase = ri + ((lane >= 16) ? 8 : 0);
    const int igBase = Ti * ST + rowBase;

    float lsum = 0.0f;
#pragma unroll
    for (int r = 0; r < 8; ++r) {
        float d2 = rowsq[rowBase + r] + sqj - 2.0f * g[r];
        d2 = fmaxf(d2, 0.0f);
        // raw v_sqrt_f32: ~1 ulp, ample for the 0.05 hinge; avoids the IEEE
        // correction + denormal-rescale chain in the dominant loop
        float v = fmaxf(MINDIST_C - __builtin_amdgcn_sqrtf(d2), 0.0f);
        lsum += (jg > igBase + r) ? v : 0.0f;  // strict upper triangle only
    }

    // wave32 shuffle reduction, then cross-wave via LDS
#pragma unroll
    for (int off = 16; off > 0; off >>= 1)
        lsum += __shfl_xor(lsum, off, 32);
    if (lane == 0) wavesum[wave] = lsum;
    __syncthreads();
    if (tid == 0) {
        float s = 0.0f;
#pragma unroll
        for (int w = 0; w < 16; ++w) s += wavesum[w];
        atomicAdd(&acc[0], s);
    }
}

// ---------------- separable gaussian blur ----------------
__global__ void blur_h_kernel(const float* __restrict__ img,
                              float* __restrict__ out) {
    size_t idx = (size_t)blockIdx.x * blockDim.x + threadIdx.x;
    if (idx >= (size_t)NB * H * W) return;
    int w = (int)(idx & (W - 1));
    size_t rowbase = idx - w;
    float s = 0.0f;
#pragma unroll
    for (int k = 0; k < KS; ++k) {
        int ww = refl511(w - PAD + k);
        s = fmaf(KW[k], img[rowbase + ww], s);
    }
    out[idx] = s;
}

__global__ void blur_v_kernel(const float* __restrict__ tmp,
                              float* __restrict__ out) {
    size_t idx = (size_t)blockIdx.x * blockDim.x + threadIdx.x;
    if (idx >= (size_t)NB * H * W) return;
    int w = (int)(idx & (W - 1));
    int h = (int)((idx >> 9) & (H - 1));
    size_t imgbase = idx - w - ((size_t)h << 9);
    float s = 0.0f;
#pragma unroll
    for (int k = 0; k < KS; ++k) {
        int hh = refl511(h - PAD + k);
        s = fmaf(KW[k], tmp[imgbase + ((size_t)hh << 9) + w], s);
    }
    out[idx] = s;
}

// ---------------- bilinear-ish intensity sampling + reduce ----------------
__global__ void intensity_kernel(const float* __restrict__ trace,
                                 const float* __restrict__ blur,
                                 float* __restrict__ acc) {
    int p = blockIdx.x * blockDim.x + threadIdx.x;   // 0 .. NB*NPTS-1
    float val = 0.0f;
    if (p < NB * NPTS) {
        int b = p >> 12;                              // / NPTS
        float t0 = trace[(size_t)p * 2 + 0];
        float t1 = trace[(size_t)p * 2 + 1];
        float idx0 = t0 * (float)H;
        float idx1 = t1 * (float)W;
        float i0 = floorf(idx0 + 0.5f);
        float j0 = floorf(idx1 + 0.5f);
        float i1 = i0 + 1.0f;
        float j1 = j0 + 1.0f;
        const float* x = blur + (size_t)b * H * W;
        int i0i = refl511((int)i0), i1i = refl511((int)i1);
        int j0i = refl511((int)j0), j1i = refl511((int)j1);
        float y00 = x[(size_t)i0i * W + j0i];
        float y10 = x[(size_t)i1i * W + j0i];
        float y01 = x[(size_t)i0i * W + j1i];
        float oa0 = (i1 - idx0) * y00 + (idx0 - i0) * y10;
        float oa1 = (j1 - idx1) * y00 + (idx1 - j0) * y01;
        val = 0.5f * (oa0 + oa1);
    }
    // wave32 shuffle reduce + cross-wave LDS
#pragma unroll
    for (int off = 16; off > 0; off >>= 1)
        val += __shfl_xor(val, off, 32);
    __shared__ float wavesum[8];
    int wave = threadIdx.x >> 5;
    int lane = threadIdx.x & 31;
    if (lane == 0) wavesum[wave] = val;
    __syncthreads();
    if (threadIdx.x == 0) {
        float s = 0.0f;
#pragma unroll
        for (int w = 0; w < 8; ++w) s += wavesum[w];
        atomicAdd(&acc[1], s);
    }
}

// ---------------- combine to scalar loss ----------------
__global__ void finalize_kernel(const float* __restrict__ acc,
                                float* __restrict__ out) {
    if (blockIdx.x == 0 && threadIdx.x == 0) {
        const float inv_total_pairs =
            1.0f / ((float)NB * ((float)NPTS * (float)(NPTS - 1) * 0.5f));
        float dist_loss = acc[0] * inv_total_pairs;
        float int_mean = acc[1] * (1.0f / (float)(NB * NPTS));
        out[0] = dist_loss + (MAXVAL_C - int_mean);
    }
}

extern "C" void kernel_launch(void* const* d_in, const int* in_sizes, int n_in,
                              void* d_out, int out_size, void* d_ws, size_t ws_size,
                              hipStream_t stream) {
    const float* trace = (const float*)d_in[0];   // [8,4096,2] f32
    const float* img   = (const float*)d_in[1];   // [8,512,512] f32
    float* out = (float*)d_out;

    // workspace layout (256B aligned slices)
    char* ws = (char*)d_ws;
    float* acc = (float*)ws;                      // 2 floats
    size_t off = 256;
    float* tmp  = (float*)(ws + off); off += (size_t)NB * H * W * sizeof(float);
    float* blur = (float*)(ws + off);

    const size_t npix = (size_t)NB * H * W;

    init_acc_kernel<<<1, 32, 0, stream>>>(acc);

    dist_wmma_kernel<<<NB * SUPER_UT, 512, 0, stream>>>(trace, acc);

    blur_h_kernel<<<(int)((npix + 255) / 256), 256, 0, stream>>>(img, tmp);
    blur_v_kernel<<<(int)((npix + 255) / 256), 256, 0, stream>>>(tmp, blur);

    intensity_kernel<<<(NB * NPTS + 255) / 256, 256, 0, stream>>>(trace, blur, acc);

    finalize_kernel<<<1, 32, 0, stream>>>(acc, out);
}